// AllPole_16200616640872
// MI455X (gfx1250) — compile-verified
//
#include <hip/hip_runtime.h>

typedef __attribute__((ext_vector_type(2))) float v2f;
typedef __attribute__((ext_vector_type(8))) float v8f;
typedef __attribute__((ext_vector_type(4))) unsigned int u32x4;
typedef __attribute__((ext_vector_type(4))) int i32x4;
typedef __attribute__((ext_vector_type(8))) int i32x8;

static constexpr int kB   = 32;    // batch (== wave32 lanes)
static constexpr int kC   = 16;    // channels
static constexpr int kT   = 16384; // time
static constexpr int kN   = 33;    // a0 + 32 feedback taps
static constexpr int kP   = 32;    // feedback order (== state size)
static constexpr int kL   = 16;    // time block (WMMA M dim)
static constexpr int kNCH = 32;    // time chunks per channel
static constexpr int kTc  = kT / kNCH;  // 512
static constexpr int kSLAB = 256;  // phase-C rows per workgroup

static constexpr int AN_PITCH = 48; // taps a1..a32 at [0..31], zero pad [32..47]
static constexpr int SW_PITCH = 65; // 64 columns (32 real + 32 homogeneous) + pad

#if __has_builtin(__builtin_amdgcn_tensor_load_to_lds) && __has_builtin(__builtin_amdgcn_s_wait_tensorcnt)
#define HAVE_TDM 1
#else
#define HAVE_TDM 0
#endif

#if HAVE_TDM
// Issue an async TDM load of a 2D tile: kN rows x kL floats, row stride kT floats,
// from global `gsrc` into LDS byte offset `lds_off`. Tracked by TENSORcnt.
__device__ inline void tdm_load_a_block(const float* gsrc, unsigned lds_off) {
    unsigned long long ga = (unsigned long long)(uintptr_t)gsrc;
    u32x4 g0;
    g0.x = 1u;                                        // count=1, user descriptor
    g0.y = lds_off;                                   // lds_addr (bytes)
    g0.z = (unsigned)(ga & 0xFFFFFFFFu);              // global_addr[31:0]
    g0.w = (unsigned)((ga >> 32) & 0x01FFFFFFu)       // global_addr[56:32]
         | (2u << 30);                                // type = 2 ("image")
    i32x8 g1;
    g1[0] = (int)(2u << 16);                          // data_size = 4 bytes
    g1[1] = (int)(((unsigned)kT & 0xFFFFu) << 16);    // tensor_dim0[15:0] @ [63:48]
    g1[2] = (int)(((unsigned)kT >> 16)                // tensor_dim0[31:16]
         | (((unsigned)kN & 0xFFFFu) << 16));         // tensor_dim1[15:0]
    g1[3] = (int)(((unsigned)kN >> 16)                // tensor_dim1[31:16]
         | ((unsigned)kL << 16));                     // tile_dim0 = 16
    g1[4] = (int)(unsigned)kN;                        // tile_dim1 = 33, tile_dim2 = 0
    g1[5] = (int)(unsigned)kT;                        // tensor_dim0_stride[31:0]
    g1[6] = 0;                                        // stride hi / dim1_stride lo
    g1[7] = 0;
    i32x4 z4 = {0, 0, 0, 0};                          // 2-D tile: groups 2/3 unused
#if defined(__clang_major__) && (__clang_major__ >= 23)
    i32x8 z8 = {0, 0, 0, 0, 0, 0, 0, 0};
    __builtin_amdgcn_tensor_load_to_lds(g0, g1, z4, z4, z8, 0);
#else
    __builtin_amdgcn_tensor_load_to_lds(g0, g1, z4, z4, 0);
#endif
}
#endif

// ---------------------------------------------------------------------------
// Phase A: per (channel, chunk): block-recursive IIR with 64 columns:
//   cols  0..31 : real batches, input x, zero initial state  -> y0 (to d_out)
//   cols 32..63 : unit initial states, zero input            -> G  (to d_ws)
// History part of each 16-step block = M(16x32) @ S(32x64) via 4 WMMA tiles.
// Coefficient tiles are DMA'd by the TDM into double-buffered LDS staging.
// ---------------------------------------------------------------------------
__global__ __launch_bounds__(32)
void allpole_phaseA(const float* __restrict__ x, const float* __restrict__ a,
                    float* __restrict__ y, float* __restrict__ G)
{
    __shared__ float AnRaw[2][kN][kL];   // TDM staging: raw a tiles (double buffered)
    __shared__ float An[kL][AN_PITCH];
    __shared__ float Sst[kP][SW_PITCH];
    __shared__ float Hs[kL][SW_PITCH];
    __shared__ float r0[kL];

    const int lane = threadIdx.x;
    const int ch   = blockIdx.x;       // time chunk
    const int c    = blockIdx.y;       // channel
    const int half = lane >> 4;
    const int l16  = lane & 15;
    const int tb   = ch * kTc;

    #pragma unroll
    for (int k = 0; k < kP; ++k) {
        Sst[k][lane]      = 0.0f;                       // real: zero init
        Sst[k][32 + lane] = (k == lane) ? 1.0f : 0.0f;  // homogeneous: identity
    }
    for (int idx = lane; idx < kL * 16; idx += 32)
        An[idx >> 4][32 + (idx & 15)] = 0.0f;           // triangular zero pad
    __syncthreads();

    const float* xch = x + ((size_t)lane * kC + c) * kT;
    float*       ych = y + ((size_t)lane * kC + c) * kT;
    const float* ach = a + (size_t)c * kN * kT;
    float*       Gch = G + (size_t)c * kT * kP;         // [t][m]

#if HAVE_TDM
    const unsigned ldsAn0 = (unsigned)(uintptr_t)(void*)&AnRaw[0][0][0];
    const unsigned ldsAn1 = (unsigned)(uintptr_t)(void*)&AnRaw[1][0][0];
    tdm_load_a_block(ach + tb, ldsAn0);                 // prime the pipeline
#endif

    int sb = 0, buf = 0;
    for (int t0 = tb; t0 < tb + kTc; t0 += kL) {
#if HAVE_TDM
        __builtin_amdgcn_s_wait_tensorcnt(0);           // current tile resident
        if (t0 + kL < tb + kTc)                         // overlap next DMA w/ compute
            tdm_load_a_block(ach + t0 + kL, buf ? ldsAn0 : ldsAn1);
#else
        for (int idx = lane; idx < kN * kL; idx += 32) {
            int j = idx >> 4, i = idx & 15;
            AnRaw[buf][j][i] = ach[(size_t)j * kT + t0 + i];
        }
#endif
        __syncthreads();
        // ---- normalize coefficients from staged tile ----
        if (lane < kL) r0[lane] = 1.0f / AnRaw[buf][0][lane];
        __syncthreads();
        #pragma unroll
        for (int q = 0; q < (kP * kL) / 32; ++q) {
            int idx = q * 32 + lane;
            int j   = (idx >> 4) + 1;
            int i   = idx & 15;
            An[i][j - 1] = AnRaw[buf][j][i] * r0[i];
        }
        if (t0 + kL < tb + kTc)
            __builtin_prefetch(xch + t0 + kL, 0, 0);
        __syncthreads();

        // ---- H = M(16x32) @ S(32x64): 4 N-tiles, K chained by 4 ----
        v8f acc0 = {}, acc1 = {}, acc2 = {}, acc3 = {};
        #pragma unroll
        for (int k0 = 0; k0 < kP; k0 += 4) {
            const int kA0 = k0 + half * 2, kA1 = kA0 + 1;
            v2f af;
            af.x = An[l16][31 + l16 - kA0];
            af.y = An[l16][31 + l16 - kA1];
            const int rA0 = (sb + kA0) & 31, rA1 = (sb + kA1) & 31;
            v2f b0, b1, b2, b3;
            b0.x = Sst[rA0][l16];      b0.y = Sst[rA1][l16];
            b1.x = Sst[rA0][16 + l16]; b1.y = Sst[rA1][16 + l16];
            b2.x = Sst[rA0][32 + l16]; b2.y = Sst[rA1][32 + l16];
            b3.x = Sst[rA0][48 + l16]; b3.y = Sst[rA1][48 + l16];
            acc0 = __builtin_amdgcn_wmma_f32_16x16x4_f32(false, af, false, b0, (short)0, acc0, false, false);
            acc1 = __builtin_amdgcn_wmma_f32_16x16x4_f32(false, af, false, b1, (short)0, acc1, false, false);
            acc2 = __builtin_amdgcn_wmma_f32_16x16x4_f32(false, af, false, b2, (short)0, acc2, false, false);
            acc3 = __builtin_amdgcn_wmma_f32_16x16x4_f32(false, af, false, b3, (short)0, acc3, false, false);
        }
        #pragma unroll
        for (int v = 0; v < 8; ++v) {
            Hs[v + half * 8][l16]      = acc0[v];
            Hs[v + half * 8][16 + l16] = acc1[v];
            Hs[v + half * 8][32 + l16] = acc2[v];
            Hs[v + half * 8][48 + l16] = acc3[v];
        }
        __syncthreads();

        // ---- triangular within-block recurrence (real + homogeneous cols) ----
        float4 xq0 = *(const float4*)(xch + t0 + 0);
        float4 xq1 = *(const float4*)(xch + t0 + 4);
        float4 xq2 = *(const float4*)(xch + t0 + 8);
        float4 xq3 = *(const float4*)(xch + t0 + 12);
        float xv[kL] = {xq0.x, xq0.y, xq0.z, xq0.w, xq1.x, xq1.y, xq1.z, xq1.w,
                        xq2.x, xq2.y, xq2.z, xq2.w, xq3.x, xq3.y, xq3.z, xq3.w};
        float yr[kL], yh[kL];
        #pragma unroll
        for (int i = 0; i < kL; ++i) {
            float tr = xv[i] - Hs[i][lane];
            float th = -Hs[i][32 + lane];
            #pragma unroll
            for (int j = 1; j <= i; ++j) {
                float aj = An[i][j - 1];
                tr = __builtin_fmaf(-aj, yr[i - j], tr);
                th = __builtin_fmaf(-aj, yh[i - j], th);
            }
            yr[i] = tr; yh[i] = th;
        }

        // ---- stores: y0 (b128) and G rows (coalesced) ----
        float4 o0 = {yr[0], yr[1], yr[2], yr[3]};
        float4 o1 = {yr[4], yr[5], yr[6], yr[7]};
        float4 o2 = {yr[8], yr[9], yr[10], yr[11]};
        float4 o3 = {yr[12], yr[13], yr[14], yr[15]};
        *(float4*)(ych + t0 + 0)  = o0;
        *(float4*)(ych + t0 + 4)  = o1;
        *(float4*)(ych + t0 + 8)  = o2;
        *(float4*)(ych + t0 + 12) = o3;
        #pragma unroll
        for (int i = 0; i < kL; ++i)
            Gch[(size_t)(t0 + i) * kP + lane] = yh[i];

        // ---- rotate circular state ----
        __syncthreads();
        #pragma unroll
        for (int i = 0; i < kL; ++i) {
            Sst[(sb + i) & 31][lane]      = yr[i];
            Sst[(sb + i) & 31][32 + lane] = yh[i];
        }
        sb = (sb + kL) & 31;
        buf ^= 1;
        __syncthreads();
    }
}

// ---------------------------------------------------------------------------
// Phase B: per channel, chain true incoming states across chunks:
//   s_in[ch+1] = y0_tail[ch] + Phi_ch @ s_in[ch],  Phi_ch = last 32 rows of G.
// lane = batch; publishes s_in[ch] (32x32, [k][b]) to d_ws for phase C.
// ---------------------------------------------------------------------------
__global__ __launch_bounds__(32)
void allpole_phaseB(const float* __restrict__ y, const float* __restrict__ G,
                    float* __restrict__ Sin)
{
    __shared__ float Phi[kP][kP + 1];
    __shared__ float Scur[kP][kB + 1];
    const int lane = threadIdx.x;
    const int c    = blockIdx.x;
    const float* ych = y + ((size_t)lane * kC + c) * kT;
    const float* Gch = G + (size_t)c * kT * kP;
    float*       Sc  = Sin + (size_t)c * kNCH * kP * kB;

    #pragma unroll
    for (int k = 0; k < kP; ++k) Scur[k][lane] = 0.0f;
    __syncthreads();

    for (int ch = 0; ch < kNCH; ++ch) {
        for (int k = 0; k < kP; ++k)
            Sc[((size_t)ch * kP + k) * kB + lane] = Scur[k][lane];
        if (ch == kNCH - 1) break;

        for (int idx = lane; idx < kP * kP; idx += 32) {
            int k = idx >> 5, m = idx & 31;
            Phi[k][m] = Gch[(size_t)((ch + 1) * kTc - kP + k) * kP + m];
        }
        __syncthreads();
        float sn[kP];
        #pragma unroll
        for (int k = 0; k < kP; ++k) {
            float acc = ych[(ch + 1) * kTc - kP + k];   // zero-state tail
            for (int m = 0; m < kP; ++m)
                acc = __builtin_fmaf(Phi[k][m], Scur[m][lane], acc);
            sn[k] = acc;
        }
        __syncthreads();
        #pragma unroll
        for (int k = 0; k < kP; ++k) Scur[k][lane] = sn[k];
        __syncthreads();
    }
}

// ---------------------------------------------------------------------------
// Phase C: y += G(Tc x 32) @ s_in(32 x 32) per (channel, chunk) — WMMA GEMM.
// ---------------------------------------------------------------------------
__global__ __launch_bounds__(32)
void allpole_phaseC(float* __restrict__ y, const float* __restrict__ G,
                    const float* __restrict__ Sin)
{
    __shared__ float Sl[kP][kB + 1];
    const int lane = threadIdx.x, half = lane >> 4, l16 = lane & 15;
    const int slab = blockIdx.x;
    const int ch   = blockIdx.y;
    const int c    = blockIdx.z;

    const float* Sc = Sin + ((size_t)c * kNCH + ch) * kP * kB;
    #pragma unroll
    for (int k = 0; k < kP; ++k) Sl[k][lane] = Sc[(size_t)k * kB + lane];
    __syncthreads();

    const float* Gch   = G + (size_t)c * kT * kP;
    const int    tbase = ch * kTc + slab * kSLAB;

    for (int tt = 0; tt < kSLAB; tt += kL) {
        const int t0 = tbase + tt;
        v8f acc0 = {}, acc1 = {};
        #pragma unroll
        for (int k0 = 0; k0 < kP; k0 += 4) {
            const int kA0 = k0 + half * 2;
            v2f af = *(const v2f*)(Gch + (size_t)(t0 + l16) * kP + kA0); // A: rows=t, K pair
            v2f b0, b1;
            b0.x = Sl[kA0][l16];      b0.y = Sl[kA0 + 1][l16];
            b1.x = Sl[kA0][16 + l16]; b1.y = Sl[kA0 + 1][16 + l16];
            acc0 = __builtin_amdgcn_wmma_f32_16x16x4_f32(false, af, false, b0, (short)0, acc0, false, false);
            acc1 = __builtin_amdgcn_wmma_f32_16x16x4_f32(false, af, false, b1, (short)0, acc1, false, false);
        }
        // D: lane l16 -> batch column, rows t0 + half*8 + v (8 consecutive t)
        float* yp0 = y + ((size_t)l16 * kC + c) * kT + t0 + half * 8;
        float4 p0 = *(const float4*)(yp0);
        float4 p1 = *(const float4*)(yp0 + 4);
        p0.x += acc0[0]; p0.y += acc0[1]; p0.z += acc0[2]; p0.w += acc0[3];
        p1.x += acc0[4]; p1.y += acc0[5]; p1.z += acc0[6]; p1.w += acc0[7];
        *(float4*)(yp0)     = p0;
        *(float4*)(yp0 + 4) = p1;
        float* yp1 = y + ((size_t)(16 + l16) * kC + c) * kT + t0 + half * 8;
        float4 q0 = *(const float4*)(yp1);
        float4 q1 = *(const float4*)(yp1 + 4);
        q0.x += acc1[0]; q0.y += acc1[1]; q0.z += acc1[2]; q0.w += acc1[3];
        q1.x += acc1[4]; q1.y += acc1[5]; q1.z += acc1[6]; q1.w += acc1[7];
        *(float4*)(yp1)     = q0;
        *(float4*)(yp1 + 4) = q1;
    }
}

extern "C" void kernel_launch(void* const* d_in, const int* in_sizes, int n_in,
                              void* d_out, int out_size, void* d_ws, size_t ws_size,
                              hipStream_t stream) {
    (void)in_sizes; (void)n_in; (void)out_size; (void)ws_size;
    const float* x = (const float*)d_in[0];  // (B, C, T)
    const float* a = (const float*)d_in[1];  // (C, N, T)
    float*       y = (float*)d_out;          // (B, C, T)
    // workspace: G = C*T*32 floats (32 MiB), Sin = C*NCH*32*32 floats (2 MiB)
    float* G   = (float*)d_ws;
    float* Sin = G + (size_t)kC * kT * kP;

    dim3 gA(kNCH, kC);
    allpole_phaseA<<<gA, 32, 0, stream>>>(x, a, y, G);
    allpole_phaseB<<<kC, 32, 0, stream>>>(y, G, Sin);
    dim3 gC(kTc / kSLAB, kNCH, kC);
    allpole_phaseC<<<gC, 32, 0, stream>>>(y, G, Sin);
}